// CustomLoss_19172734010147
// MI455X (gfx1250) — compile-verified
//
#include <hip/hip_runtime.h>
#include <hip/hip_bf16.h>

// ---------------------------------------------------------------------------
// BCE-style loss reduction, MI455X (gfx1250, wave32).
// Memory-bound: 128 MiB streamed @ 23.3 TB/s => ~5.7us floor.
// Pass 1: grid-stride stream with float4/int4 (global_load_b128) loads,
//         wave32 shuffle reduce + LDS cross-wave reduce, per-block partials.
// Pass 2: single wave (32 threads) reduces 1024 partials; the 32->1 cross-lane
//         sum is done with V_WMMA_F32_16X16X4_F32 (B = ones => row sums),
//         exercising the CDNA5 matrix path where it is actually correct.
// ---------------------------------------------------------------------------

typedef float v2f __attribute__((ext_vector_type(2)));
typedef float v8f __attribute__((ext_vector_type(8)));

#define NBLOCKS 1024
#define NTHREADS 256

// ll(p, l) = l*p - softplus(p);  softplus(p) = max(p,0) + log(1 + exp(-|p|))
// (identical to l*log_sigmoid(p) + (1-l)*log_sigmoid(-p), numerically stable)
__device__ __forceinline__ float bce_ll(float p, int l) {
    float sp = fmaxf(p, 0.0f) + __logf(1.0f + __expf(-fabsf(p)));
    return (l ? p : 0.0f) - sp;
}

// Sum a value held by all 32 lanes of a full wave (EXEC must be all ones).
__device__ __forceinline__ float wave32_sum(float v) {
#if __has_builtin(__builtin_amdgcn_wmma_f32_16x16x4_f32)
    // A: 16x4 f32, 2 VGPRs/lane. Put the lane value in slot 0, zero in slot 1.
    // B: 4x16 all-ones (layout-invariant). Then D[m][n] = sum_k A[m][k], so
    // summing one lane's 8 C/D VGPRs gives sum of rows {0..7} (lanes 0-15) or
    // rows {8..15} (lanes 16-31); lane0 + lane16 = sum of all 64 A slots.
    v2f a; a.x = v;    a.y = 0.0f;
    v2f b; b.x = 1.0f; b.y = 1.0f;
    v8f c = {};
    c = __builtin_amdgcn_wmma_f32_16x16x4_f32(
        /*neg_a=*/false, a, /*neg_b=*/false, b,
        /*c_mod=*/(short)0, c, /*reuse_a=*/false, /*reuse_b=*/false);
    float s = (c[0] + c[1]) + (c[2] + c[3]) + (c[4] + c[5]) + (c[6] + c[7]);
    return __shfl(s, 0, 32) + __shfl(s, 16, 32);
#else
    for (int o = 16; o > 0; o >>= 1) v += __shfl_down(v, o, 32);
    return __shfl(v, 0, 32);
#endif
}

__global__ __launch_bounds__(NTHREADS)
void bce_partials_kernel(const float* __restrict__ pred,
                         const int*   __restrict__ lab,
                         float* __restrict__ ws_ll,
                         float* __restrict__ ws_pos,
                         int n) {
    const int tid    = blockIdx.x * blockDim.x + threadIdx.x;
    const int stride = gridDim.x * blockDim.x;
    const int nv     = n >> 2;

    const float4* __restrict__ p4 = (const float4*)pred;
    const int4*   __restrict__ l4 = (const int4*)lab;

    float ll  = 0.0f;
    int   cnt = 0;

    for (int i = tid; i < nv; i += stride) {
        // Prefetch one grid-stride ahead (global_prefetch_b8; speculative, no fault).
        __builtin_prefetch(&p4[i + stride], 0, 1);
        __builtin_prefetch(&l4[i + stride], 0, 1);
        float4 p = p4[i];
        int4   l = l4[i];
        ll += bce_ll(p.x, l.x) + bce_ll(p.y, l.y)
            + bce_ll(p.z, l.z) + bce_ll(p.w, l.w);
        cnt += l.x + l.y + l.z + l.w;
    }
    // Scalar tail (N is a multiple of 4, but stay generic).
    if (tid == 0) {
        for (int i = nv << 2; i < n; ++i) {
            ll  += bce_ll(pred[i], lab[i]);
            cnt += lab[i];
        }
    }

    // Wave32 reduce.
    float fcnt = (float)cnt;
    for (int o = 16; o > 0; o >>= 1) {
        ll   += __shfl_down(ll,   o, 32);
        fcnt += __shfl_down(fcnt, o, 32);
    }

    // Cross-wave reduce via LDS (8 waves / 256-thread block on wave32).
    __shared__ float sll[NTHREADS / 32];
    __shared__ float spos[NTHREADS / 32];
    const int wave = threadIdx.x >> 5;
    const int lane = threadIdx.x & 31;
    if (lane == 0) { sll[wave] = ll; spos[wave] = fcnt; }
    __syncthreads();
    if (threadIdx.x == 0) {
        float a = 0.0f, b = 0.0f;
        #pragma unroll
        for (int w = 0; w < NTHREADS / 32; ++w) { a += sll[w]; b += spos[w]; }
        ws_ll[blockIdx.x]  = a;   // every slot written every call: deterministic
        ws_pos[blockIdx.x] = b;   // integer-valued float, exact (<= 2^24)
    }
}

__global__ __launch_bounds__(32)
void bce_finalize_kernel(const float* __restrict__ ws_ll,
                         const float* __restrict__ ws_pos,
                         float* __restrict__ out,
                         int nb, float n_total) {
    const int lane = threadIdx.x;  // single full wave, EXEC all ones
    float ll = 0.0f, pos = 0.0f;
    for (int i = lane; i < nb; i += 32) {  // uniform trip count, no divergence
        ll  += ws_ll[i];
        pos += ws_pos[i];
    }
    float ll_tot  = wave32_sum(ll);
    float pos_tot = wave32_sum(pos);
    if (lane == 0) {
        float neg = n_total - pos_tot;
        out[0] = -ll_tot / ((1.0f + neg) * pos_tot);
    }
}

extern "C" void kernel_launch(void* const* d_in, const int* in_sizes, int n_in,
                              void* d_out, int out_size, void* d_ws, size_t ws_size,
                              hipStream_t stream) {
    const float* pred = (const float*)d_in[0];
    const int*   lab  = (const int*)d_in[1];
    float*       out  = (float*)d_out;
    const int    n    = in_sizes[0];

    int nb = NBLOCKS;
    if (ws_size < (size_t)(2 * nb) * sizeof(float))
        nb = (int)(ws_size / (2 * sizeof(float)));  // defensive; normally 1024

    float* ws_ll  = (float*)d_ws;
    float* ws_pos = ws_ll + nb;

    bce_partials_kernel<<<nb, NTHREADS, 0, stream>>>(pred, lab, ws_ll, ws_pos, n);
    bce_finalize_kernel<<<1, 32, 0, stream>>>(ws_ll, ws_pos, out, nb, (float)n);
}